// BitLevelMapper_18691697672519
// MI455X (gfx1250) — compile-verified
//
#include <hip/hip_runtime.h>
#include <stdint.h>

// BitLevelMapper forward, CUMULATIVE context + FLIP output.
// Roofline: 256MB in + 256MB out => ~22us at 23.3 TB/s; pure streaming.
// Row i of the table only uses addresses < 2^i and holds {0,1}, so the whole
// effective LUT bit-packs into 2052 dwords (8.2 KB):
//   kernel 1 packs it into d_ws (one word/thread, 8 unrolled b128 loads),
//   kernel 2 async-stages it into LDS (CDNA5 ASYNCcnt path) and streams the
//   512MB with nontemporal b128 accesses; rows 0..5 resolved via scalar loads.

typedef int   v4i __attribute__((ext_vector_type(4)));
typedef float v4f __attribute__((ext_vector_type(4)));

#define N_BITS            16
#define TABLE_SIZE        32768
#define PACKED_WORDS      2052   // sum_i max(1, 2^(i-5)), i=0..15
#define PACKED_WORDS_PAD  2064   // multiple of 4 -> clean b128 async staging
#define BLOCK             256
#define EPT               4      // elements per thread in main kernel

// Word offset of packed row i: i for i<6, else 4 + 2^(i-5).
__device__ static constexpr int kBase[16] =
    {0, 1, 2, 3, 4, 5, 6, 8, 12, 20, 36, 68, 132, 260, 516, 1028};

// ---------------------------------------------------------------------------
// Kernel 2 (defined first so the disasm snippet shows the async staging):
// main streaming pass, EPT elements per thread.
// ---------------------------------------------------------------------------
__global__ void __launch_bounds__(BLOCK)
bitmapper_kernel(const int* __restrict__ bits,
                 const uint32_t* __restrict__ packed_g,
                 float* __restrict__ out, int n_elems) {
  __shared__ uint32_t s_packed[PACKED_WORDS_PAD];  // 8256 B

  // ---- Stage packed LUT global->LDS via CDNA5 async copies (ASYNCcnt). ----
  for (int idx = threadIdx.x; idx < PACKED_WORDS_PAD / 4; idx += BLOCK) {
    uint32_t lds_addr = (uint32_t)(uintptr_t)(&s_packed[idx * 4]);
    uint64_t gaddr    = (uint64_t)(uintptr_t)(packed_g + idx * 4);
    asm volatile("global_load_async_to_lds_b128 %0, %1, off"
                 :: "v"(lds_addr), "v"(gaddr) : "memory");
  }

  // Rows 0..5 are one uniform word each -> scalar loads (SGPR broadcast).
  uint32_t u0 = packed_g[0], u1 = packed_g[1], u2 = packed_g[2];
  uint32_t u3 = packed_g[3], u4 = packed_g[4], u5 = packed_g[5];

  asm volatile("s_wait_asynccnt 0" ::: "memory");
  __syncthreads();

  const int e0 = blockIdx.x * (BLOCK * EPT) + threadIdx.x;

#pragma unroll
  for (int r = 0; r < EPT; ++r) {
    const int e = e0 + r * BLOCK;
    if (e >= n_elems) continue;

    // ---- Read 16 input bits as 4x b128 nontemporal loads. ----
    const v4i* bp = reinterpret_cast<const v4i*>(bits) + (size_t)e * 4;
    v4i b0 = __builtin_nontemporal_load(bp + 0);
    v4i b1 = __builtin_nontemporal_load(bp + 1);
    v4i b2 = __builtin_nontemporal_load(bp + 2);
    v4i b3 = __builtin_nontemporal_load(bp + 3);

    // packed value p: bit k = bits_rev[k] = bits[15-k]
    uint32_t p = 0;
#pragma unroll
    for (int j = 0; j < 4; ++j) {
      p |= ((uint32_t)b0[j] & 1u) << (15 - j);
      p |= ((uint32_t)b1[j] & 1u) << (11 - j);
      p |= ((uint32_t)b2[j] & 1u) << (7 - j);
      p |= ((uint32_t)b3[j] & 1u) << (3 - j);
    }

    // ---- 16 single-bit lookups: rows 0..5 from SGPRs, 6..15 from LDS. ----
    const uint32_t uw[6] = {u0, u1, u2, u3, u4, u5};
    float o[16];
#pragma unroll
    for (int i = 0; i < N_BITS; ++i) {
      const uint32_t addr = p & ((1u << i) - 1u);  // low i bits
      uint32_t word;
      if (i < 6) word = uw[i];                     // i constant: folds to SGPR
      else       word = s_packed[(uint32_t)kBase[i] + (addr >> 5)];
      const uint32_t flip = (word >> (addr & 31u)) & 1u;
      const uint32_t bit  = (p >> i) & 1u;         // bits_rev[i]
      o[15 - i] = (float)(bit ^ flip);             // out[15-i] = out_rev[i]
    }

    // ---- Write 16 floats as 4x b128 nontemporal stores. ----
    v4f* op = reinterpret_cast<v4f*>(out) + (size_t)e * 4;
#pragma unroll
    for (int q = 0; q < 4; ++q) {
      v4f w;
      w[0] = o[4 * q + 0]; w[1] = o[4 * q + 1];
      w[2] = o[4 * q + 2]; w[3] = o[4 * q + 3];
      __builtin_nontemporal_store(w, op + q);
    }
  }
}

// ---------------------------------------------------------------------------
// Kernel 1: bit-pack the effective (prefix) table into d_ws.
// One packed word per thread; 8 unrolled b128 loads so all 32 element loads
// are in flight simultaneously (no serial latency chain).
// ---------------------------------------------------------------------------
__global__ void __launch_bounds__(BLOCK)
pack_tables_kernel(const float* __restrict__ tables,
                   uint32_t* __restrict__ packed) {
  const int w = blockIdx.x * BLOCK + threadIdx.x;
  if (w >= PACKED_WORDS_PAD) return;

  uint32_t word = 0u;
  if (w < PACKED_WORDS) {
    // Row-finding via clz: w<6 -> row=w, sub=0; else row = 5 + log2(w-4).
    int row, sub;
    if (w < 6) { row = w; sub = 0; }
    else {
      const uint32_t t = (uint32_t)(w - 4);
      row = 5 + (31 - __builtin_clz(t));
      sub = w - (4 + (1 << (row - 5)));
    }
    // Always read 32 floats (in-bounds for every row), mask invalid bits.
    const v4f* src = reinterpret_cast<const v4f*>(
        tables + (size_t)row * TABLE_SIZE + (size_t)sub * 32);
#pragma unroll
    for (int q = 0; q < 8; ++q) {
      v4f f = __builtin_nontemporal_load(src + q);
#pragma unroll
      for (int c = 0; c < 4; ++c)
        word |= (f[c] > 0.5f ? 1u : 0u) << (4 * q + c);
    }
    const uint32_t mask =
        (row < 5) ? ((1u << (1 << row)) - 1u) : 0xFFFFFFFFu;
    word &= mask;
  }
  packed[w] = word;  // pad words written as 0 so async staging reads init'd mem
}

// ---------------------------------------------------------------------------
extern "C" void kernel_launch(void* const* d_in, const int* in_sizes, int n_in,
                              void* d_out, int out_size, void* d_ws, size_t ws_size,
                              hipStream_t stream) {
  const int*   bits   = (const int*)d_in[0];     // [B,16] int32 in {0,1}
  const float* tables = (const float*)d_in[1];   // [16, 32768] float32 in {0,1}
  float*       out    = (float*)d_out;           // [B,16] float32
  uint32_t*    packed = (uint32_t*)d_ws;         // needs 8256 B of scratch

  const int n = in_sizes[0] / N_BITS;            // B = 4194304

  const int pack_grid = (PACKED_WORDS_PAD + BLOCK - 1) / BLOCK;  // 9 blocks
  hipLaunchKernelGGL(pack_tables_kernel, dim3(pack_grid), dim3(BLOCK), 0,
                     stream, tables, packed);

  const int grid = (n + BLOCK * EPT - 1) / (BLOCK * EPT);        // 4096 blocks
  hipLaunchKernelGGL(bitmapper_kernel, dim3(grid), dim3(BLOCK), 0, stream,
                     bits, packed, out, n);
}